// SGPNLoss_15650860827299
// MI455X (gfx1250) — compile-verified
//
#include <hip/hip_runtime.h>

typedef __attribute__((ext_vector_type(2))) float v2f;
typedef __attribute__((ext_vector_type(8))) float v8f;

namespace {
constexpr int kB = 4;
constexpr int kC = 64;
constexpr int kN = 4096;
constexpr float kAlpha  = 2.0f;
constexpr float kMargin = 0.8f;
constexpr int kTilesPerRow   = kN / 16;                    // 256
constexpr int kTilesPerBatch = kTilesPerRow * kTilesPerRow; // 65536
constexpr int kWavesPerBlock = 8;                           // 256 threads, wave32
}

// --- Kernel 1: r[b,n] = sum_c Fsim[b,c,n]^2 ------------------------------
__global__ __launch_bounds__(256) void sgpn_rownorm_kernel(
    const float* __restrict__ F, float* __restrict__ r) {
  const int idx = blockIdx.x * blockDim.x + threadIdx.x;   // [0, B*N)
  const int b = idx / kN;
  const int n = idx % kN;
  const float* Fb = F + (size_t)b * kC * kN;
  float s = 0.0f;
#pragma unroll 8
  for (int c = 0; c < kC; ++c) {
    const float v = Fb[(size_t)c * kN + n];
    s += v * v;
  }
  r[idx] = s;
}

// --- Kernel 2: fused gram-GEMM (fp32 WMMA) + SGPN loss epilogue ----------
__global__ __launch_bounds__(256) void sgpn_loss_kernel(
    const float* __restrict__ F, const int* __restrict__ tgt,
    const float* __restrict__ r, double* __restrict__ acc) {
  const int lane = threadIdx.x & 31;
  const int wave = threadIdx.x >> 5;
  const int tile = blockIdx.x * kWavesPerBlock + wave;     // global 16x16 tile id
  const int b  = tile >> 16;                               // kTilesPerBatch == 65536
  const int t  = tile & (kTilesPerBatch - 1);
  const int i0 = (t >> 8) << 4;                            // tile row origin
  const int j0 = (t & 255) << 4;                           // tile col origin

  const float* Fb = F + (size_t)b * kC * kN;
  const int m16   = lane & 15;
  const int khalf = (lane < 16) ? 0 : 2;                   // ISA A/B f32 frag layout
  const int rowA  = i0 + m16;                              // A: M = lane%16
  const int colB  = j0 + m16;                              // B: N = lane%16

  v8f c = {0.f, 0.f, 0.f, 0.f, 0.f, 0.f, 0.f, 0.f};
#pragma unroll
  for (int kk = 0; kk < kC; kk += 4) {
    const float* pk = Fb + (size_t)(kk + khalf) * kN;
    v2f a, bm;
    a.x  = pk[rowA];           // A[M=rowA, K=kk+khalf]
    a.y  = pk[kN + rowA];      // A[M=rowA, K=kk+khalf+1]
    bm.x = pk[colB];           // B[K=kk+khalf,   N=colB]
    bm.y = pk[kN + colB];      // B[K=kk+khalf+1, N=colB]
    // D = A(16x4) * B(4x16) + C   (fp32 WMMA, exact vs reference)
    c = __builtin_amdgcn_wmma_f32_16x16x4_f32(
        /*neg_a=*/false, a, /*neg_b=*/false, bm,
        /*c_mod=*/(short)0, c, /*reuse_a=*/false, /*reuse_b=*/false);
  }

  // Epilogue over this 16x16 gram tile. C/D layout: VGPR g holds row
  // M = g (lanes 0-15) or g+8 (lanes 16-31), col N = lane%16.
  const float rj  = r[b * kN + colB];
  const int   tj  = tgt[b * kN + colB];
  const float tjf = (float)tj;
  const int mbase = i0 + ((lane < 16) ? 0 : 8);

  float lsum = 0.0f;
#pragma unroll
  for (int g = 0; g < 8; ++g) {
    const int   row = mbase + g;
    const float ri  = r[b * kN + row];
    const int   ti  = tgt[b * kN + row];
    float D = ri + rj - 2.0f * c[g];
    D = fmaxf(D, 0.0f);
    const bool  same = (row == colB) || ((ti >= 0) && (ti == tj));
    const float val  = same ? D : kAlpha * fmaxf(kMargin - D, 0.0f);
    lsum += val * ((float)ti * tjf);
  }

  // wave32 tree reduction, then one double atomic per wave
  double dsum = (double)lsum;
#pragma unroll
  for (int off = 16; off > 0; off >>= 1)
    dsum += __shfl_down(dsum, off, 32);
  if (lane == 0) atomicAdd(acc, dsum);
}

// --- Kernel 3: mean + write output ---------------------------------------
__global__ void sgpn_finalize_kernel(const double* __restrict__ acc,
                                     float* __restrict__ out) {
  out[0] = (float)(acc[0] * (1.0 / ((double)kB * (double)kN * (double)kN)));
}

extern "C" void kernel_launch(void* const* d_in, const int* in_sizes, int n_in,
                              void* d_out, int out_size, void* d_ws, size_t ws_size,
                              hipStream_t stream) {
  // Inputs per reference setup_inputs(): [0]=l0_points (unused by the loss),
  // [1]=Fsim float32 [B,C,N], [2]=target int32 [B,N]. Output: 1 float.
  const float* F   = (const float*)d_in[1];
  const int*   tgt = (const int*)d_in[2];

  double* acc = (double*)d_ws;                       // 8-byte accumulator
  float*  r   = (float*)((char*)d_ws + 16);          // B*N row norms (64 KB)

  hipMemsetAsync(d_ws, 0, 16, stream);               // capture-safe zeroing

  sgpn_rownorm_kernel<<<(kB * kN) / 256, 256, 0, stream>>>(F, r);

  const int tiles = kB * kTilesPerBatch;             // 262144 wave-tiles
  sgpn_loss_kernel<<<tiles / kWavesPerBlock, 256, 0, stream>>>(F, tgt, r, acc);

  sgpn_finalize_kernel<<<1, 1, 0, stream>>>(acc, (float*)d_out);

  (void)d_in; (void)in_sizes; (void)n_in; (void)out_size; (void)ws_size;
}